// ClassificationLoss_7730941132975
// MI455X (gfx1250) — compile-verified
//
#include <hip/hip_runtime.h>
#include <math.h>

// ---------------------------------------------------------------------------
// ClassificationLoss (BCE-with-logits mean + recursive tree regularization)
// MI455X / gfx1250, wave32. Bandwidth-bound (~670 MB traffic -> ~29us roofline
// at 23.3 TB/s).
//   stage 1: BCE partials, non-temporal b128 streams (protect 192MB L2)
//   stage 2: edge partials, gathered W rows (67MB -> L2-resident, RT policy)
//   stage 3: one-wave final reduce via V_WMMA_F32_16X16X4_F32 (lossless f32)
// Deterministic: fixed reduction trees, no float atomics.
// ---------------------------------------------------------------------------

#define NT       256   // threads per block (8 wave32s)
#define NB_BCE   2048  // BCE partial blocks
#define NB_REG   1024  // edge partial blocks
#define HIDDEN   1024  // row length of recursive_params

typedef __attribute__((ext_vector_type(2))) float v2f;
typedef __attribute__((ext_vector_type(4))) float v4f;  // native vec: NT-load OK
typedef __attribute__((ext_vector_type(8))) float v8f;

__device__ __forceinline__ float bce_elem(float x, float y) {
    // logaddexp(0,x) - x*y = max(x,0) + log1p(exp(-|x|)) - x*y  (stable)
    return fmaxf(x, 0.0f) + log1pf(__expf(-fabsf(x))) - x * y;
}

// ---- stage 1: BCE partial sums --------------------------------------------
__global__ __launch_bounds__(NT) void bce_partial_kernel(
    const float* __restrict__ logits,
    const float* __restrict__ targets,
    float* __restrict__ partial,
    int n4)  // number of float4 elements
{
    __shared__ float red[NT];
    const v4f* lg = (const v4f*)logits;
    const v4f* tg = (const v4f*)targets;

    int tid    = blockIdx.x * NT + threadIdx.x;
    int stride = gridDim.x * NT;

    float acc = 0.0f;
    // n4 = 16,777,216 ; stride = 524,288 -> exactly 32 iters, unrolled x2.
    for (int i = tid; i < n4; i += 2 * stride) {
        // streamed once: non-temporal so L2 stays free for stage-2 W rows
        v4f x0 = __builtin_nontemporal_load(&lg[i]);
        v4f y0 = __builtin_nontemporal_load(&tg[i]);
        int j = i + stride;
        v4f x1 = {}, y1 = {};
        bool ok = (j < n4);
        if (ok) {
            x1 = __builtin_nontemporal_load(&lg[j]);
            y1 = __builtin_nontemporal_load(&tg[j]);
        }
        acc += bce_elem(x0.x, y0.x) + bce_elem(x0.y, y0.y)
             + bce_elem(x0.z, y0.z) + bce_elem(x0.w, y0.w);
        if (ok) {
            acc += bce_elem(x1.x, y1.x) + bce_elem(x1.y, y1.y)
                 + bce_elem(x1.z, y1.z) + bce_elem(x1.w, y1.w);
        }
    }

    red[threadIdx.x] = acc;
    __syncthreads();
    for (int s = NT / 2; s > 0; s >>= 1) {
        if (threadIdx.x < s) red[threadIdx.x] += red[threadIdx.x + s];
        __syncthreads();
    }
    if (threadIdx.x == 0)
        __builtin_nontemporal_store(red[0], &partial[blockIdx.x]);
}

// ---- stage 2: recursive-regularization partial sums -----------------------
// Each block processes whole edges: 256 threads x float4 == one 1024-f32 row.
// W is 67 MB -> resident in the 192 MB L2 after first touch; default RT hint.
__global__ __launch_bounds__(NT) void reg_partial_kernel(
    const float* __restrict__ W,
    const int* __restrict__ parent_idx,
    const int* __restrict__ child_idx,
    float* __restrict__ partial,
    int n_edges)
{
    __shared__ float red[NT];
    float acc = 0.0f;

    for (int e = blockIdx.x; e < n_edges; e += gridDim.x) {
        long long p = (long long)parent_idx[e];  // uniform per block (s_load)
        long long c = (long long)child_idx[e];
        const v4f* wp = (const v4f*)(W + p * HIDDEN);
        const v4f* wc = (const v4f*)(W + c * HIDDEN);
        v4f a = wp[threadIdx.x];  // global_load_b128
        v4f b = wc[threadIdx.x];  // global_load_b128
        float dx = a.x - b.x, dy = a.y - b.y, dz = a.z - b.z, dw = a.w - b.w;
        acc = fmaf(dx, dx, acc);
        acc = fmaf(dy, dy, acc);
        acc = fmaf(dz, dz, acc);
        acc = fmaf(dw, dw, acc);
    }

    red[threadIdx.x] = acc;
    __syncthreads();
    for (int s = NT / 2; s > 0; s >>= 1) {
        if (threadIdx.x < s) red[threadIdx.x] += red[threadIdx.x + s];
        __syncthreads();
    }
    if (threadIdx.x == 0)
        __builtin_nontemporal_store(red[0], &partial[blockIdx.x]);
}

// ---- stage 3: single-wave final reduce via V_WMMA_F32_16X16X4_F32 ---------
// 32 lanes x 2 accumulators = 64 values = the 16x4 f32 A-matrix. B = ones
// (4x16) -> D[m][n] = rowsum(A[m]). Lanes 0-15 hold rows 0-7 in their 8 D
// VGPRs, lanes 16-31 hold rows 8-15; 8 adds + shfl_xor(16) = grand total.
// nb and nr are multiples of 32, so EXEC stays all-1s (WMMA requirement).
__global__ void final_reduce_kernel(
    const float* __restrict__ partial_bce, int nb, float inv_n,
    const float* __restrict__ partial_reg, int nr, float reg_scale,
    float* __restrict__ out)
{
    int lane = threadIdx.x;  // 0..31, one wave32

    float acc_bce = 0.0f;
    for (int i = lane; i < nb; i += 32) acc_bce += partial_bce[i];
    float acc_reg = 0.0f;
    for (int i = lane; i < nr; i += 32) acc_reg += partial_reg[i];

    v2f a;
    a[0] = acc_bce * inv_n;     // A-matrix VGPR0 (K even plane)
    a[1] = acc_reg * reg_scale; // A-matrix VGPR1 (K odd plane)
    v2f b;
    b[0] = 1.0f;
    b[1] = 1.0f;
    v8f c = {};

    v8f d = __builtin_amdgcn_wmma_f32_16x16x4_f32(
        /*neg_a=*/false, a, /*neg_b=*/false, b,
        /*c_mod=*/(short)0, c, /*reuse_a=*/false, /*reuse_b=*/false);

    float s = d[0] + d[1] + d[2] + d[3] + d[4] + d[5] + d[6] + d[7];
    s += __shfl_xor(s, 16, 32);  // rows 0-7 half + rows 8-15 half

    if (lane == 0) out[0] = s;
}

// ---------------------------------------------------------------------------
extern "C" void kernel_launch(void* const* d_in, const int* in_sizes, int n_in,
                              void* d_out, int out_size, void* d_ws, size_t ws_size,
                              hipStream_t stream) {
    const float* logits  = (const float*)d_in[0];
    const float* targets = (const float*)d_in[1];
    const float* W       = (const float*)d_in[2];
    const int*   pidx    = (const int*)d_in[3];
    const int*   cidx    = (const int*)d_in[4];
    float*       out     = (float*)d_out;

    int n_total = in_sizes[0];       // BATCH * N_LABELS = 67,108,864
    int n4      = n_total >> 2;      // float4 count
    int n_edges = in_sizes[3];       // N_LABELS - 1 = 16,383

    float* partial_bce = (float*)d_ws;            // NB_BCE floats
    float* partial_reg = partial_bce + NB_BCE;    // NB_REG floats (12 KB total)

    bce_partial_kernel<<<NB_BCE, NT, 0, stream>>>(logits, targets, partial_bce, n4);
    reg_partial_kernel<<<NB_REG, NT, 0, stream>>>(W, pidx, cidx, partial_reg, n_edges);

    float inv_n     = 1.0f / (float)n_total;
    float reg_scale = 0.5f * 1e-6f;  // 0.5 * RECURSIVE_PENALTY
    final_reduce_kernel<<<1, 32, 0, stream>>>(partial_bce, NB_BCE, inv_n,
                                              partial_reg, NB_REG, reg_scale,
                                              out);
}